// MemoryEfficientCrossAttention_33363305955540
// MI455X (gfx1250) — compile-verified
//
#include <hip/hip_runtime.h>

typedef __attribute__((ext_vector_type(2))) float v2f;
typedef __attribute__((ext_vector_type(4))) float v4f;
typedef __attribute__((ext_vector_type(8))) float v8f;
typedef __attribute__((ext_vector_type(4))) unsigned int v4u;
typedef __attribute__((ext_vector_type(4))) int v4i;
typedef __attribute__((ext_vector_type(8))) int v8i;

#define S_DIM 1024
#define H_DIM 1024
#define BATCH 64
#define TILE_N 128
#define QS_STRIDE 1028   // 1024 + 4 floats: conflict-free A-fragment b64 reads
#define KS_STRIDE 132    // 128 + 4 floats: matches TDM pad (interval 128 dw, pad 4 dw)
#define PS_STRIDE 132
#define OS_STRIDE 1028

// LDS partition (in floats)
#define QS_OFF   0
#define OS_OFF   (QS_OFF + 16 * QS_STRIDE)        // 16448
#define KS0_OFF  (OS_OFF + 16 * OS_STRIDE)        // 32896
#define KS1_OFF  (KS0_OFF + 128 * KS_STRIDE)      // 49792
#define PS_OFF   (KS1_OFF + 128 * KS_STRIDE)      // 66688
#define MRUN_OFF (PS_OFF + 16 * PS_STRIDE)        // 68800
#define LRUN_OFF (MRUN_OFF + 16)
#define SCL_OFF  (LRUN_OFF + 16)
#define MNEW_OFF (SCL_OFF + 16)
#define WRED_OFF (MNEW_OFF + 16)
#define LDS_FLOATS (WRED_OFF + 128)               // 68992 floats
#define LDS_BYTES  (LDS_FLOATS * 4)               // 275968 bytes (< 320KB WGP)

#define KS0_BYTE (KS0_OFF * 4)
#define KS1_BYTE (KS1_OFF * 4)

#define NEG_BIG (-3.0e38f)

// Issue one TDM load of a 128x128 f32 tile (row stride H_DIM in memory) into LDS
// at lds_byte_off, writing rows with a 4-DWORD pad after every 128 DWORDs so the
// LDS image has row stride KS_STRIDE floats. D# layout per CDNA5 ISA ch.8.
__device__ __forceinline__ void tdm_load_tile(unsigned int lds_byte_off,
                                              const float* gptr) {
    unsigned long long ga = (unsigned long long)(size_t)gptr;
    v4u g0;
    g0[0] = 1u;                                    // count=1, user mode, no gather
    g0[1] = lds_byte_off;                          // lds_addr (bytes)
    g0[2] = (unsigned int)(ga & 0xffffffffu);      // global_addr[31:0]
    g0[3] = (unsigned int)((ga >> 32) & 0x01ffffffu) | (2u << 30); // addr[56:32], type=2
    v8i g1;
    g1[0] = (2 << 16)      // data_size = 4 bytes
          | (1 << 20)      // pad_enable
          | (6 << 22)      // pad_interval: 128 DWORDs
          | (3 << 25);     // pad_amount: 4 DWORDs
    g1[1] = (H_DIM & 0xffff) << 16;                // tensor_dim0[15:0]
    g1[2] = (S_DIM & 0xffff) << 16;                // dim0 hi=0 | tensor_dim1[15:0]
    g1[3] = (128 << 16);                           // dim1 hi=0 | tile_dim0=128
    g1[4] = 128;                                   // tile_dim1=128, tile_dim2=0
    g1[5] = H_DIM;                                 // tensor_dim0_stride[31:0]
    g1[6] = 0;
    g1[7] = 0;
    v4i zz = {0, 0, 0, 0};
#if __clang_major__ >= 23
    v8i z8 = {0, 0, 0, 0, 0, 0, 0, 0};
    __builtin_amdgcn_tensor_load_to_lds(g0, g1, zz, zz, z8, 0);
#else
    __builtin_amdgcn_tensor_load_to_lds(g0, g1, zz, zz, 0);
#endif
}

__global__ __launch_bounds__(256)
void fa_pairswap_kernel(const float* __restrict__ x,
                        const int* __restrict__ mask,
                        float* __restrict__ out) {
    extern __shared__ float lds[];
    float* Qs   = lds + QS_OFF;
    float* Os   = lds + OS_OFF;
    float* Ks0  = lds + KS0_OFF;  // TDM-staged K/V chunk, buffer 0
    float* Ks1  = lds + KS1_OFF;  // TDM-staged K/V chunk, buffer 1
    float* Ps   = lds + PS_OFF;   // probability tile 16 x 128
    float* mrun = lds + MRUN_OFF;
    float* lrun = lds + LRUN_OFF;
    float* scl  = lds + SCL_OFF;
    float* mnew = lds + MNEW_OFF;
    float* wred = lds + WRED_OFF; // 8 waves x 16 rows

    const int tid  = threadIdx.x;
    const int wave = tid >> 5;
    const int lane = tid & 31;
    const int half = lane >> 4;
    const int l15  = lane & 15;

    const int qrow0   = blockIdx.x * 16;
    const int b       = blockIdx.y;
    const int partner = b ^ 1;                    // pair (p, q) <-> (p, 1-q)
    const float* Qg = x + (size_t)b * S_DIM * H_DIM;
    const float* Kg = x + (size_t)partner * S_DIM * H_DIM;
    const int*   mg = mask + (size_t)partner * S_DIM;
    float*       Og = out + (size_t)b * S_DIM * H_DIM;

    // ---- init: load Q tile, zero O, init softmax stats ----
    for (int i = 0; i < 16; ++i) {
        int r  = i;
        int c4 = tid;                             // 256 float4 per row
        v4f v = *(const v4f*)(Qg + (size_t)(qrow0 + r) * H_DIM + c4 * 4);
        *(v4f*)(Qs + r * QS_STRIDE + c4 * 4) = v;
        v4f z = {0.f, 0.f, 0.f, 0.f};
        *(v4f*)(Os + r * OS_STRIDE + c4 * 4) = z;
    }
    if (tid < 16) { mrun[tid] = NEG_BIG; lrun[tid] = 0.f; }
    __syncthreads();

    for (int T = 0; T < S_DIM; T += TILE_N) {
        const float* Kblk = Kg + (size_t)T * H_DIM;   // rows T..T+127

        // ================= Phase A: scores = Q . K^T =================
        if (wave == 0) tdm_load_tile(KS0_BYTE, Kblk);     // chunk 0 in flight
        v8f c = {0.f, 0.f, 0.f, 0.f, 0.f, 0.f, 0.f, 0.f};
        for (int hi = 0; hi < 8; ++hi) {
            const float* Kb = (hi & 1) ? Ks1 : Ks0;
            __builtin_amdgcn_s_wait_tensorcnt(0);   // issuer: chunk hi arrived
            __syncthreads();                        // visible to all waves
            if (wave == 0 && hi < 7)                // overlap: fetch chunk hi+1
                tdm_load_tile((hi & 1) ? KS0_BYTE : KS1_BYTE,
                              Kblk + (hi + 1) * 128);
            const float* arow = Qs + l15 * QS_STRIDE + hi * 128 + 2 * half;
            const float* brow = Kb + (wave * 16 + l15) * KS_STRIDE + 2 * half;
#pragma unroll
            for (int k4 = 0; k4 < 128; k4 += 4) {
                v2f a  = *(const v2f*)(arow + k4);   // A[M=l15][K]
                v2f bf = *(const v2f*)(brow + k4);   // B[K][N=l15] = K[t][h]
                c = __builtin_amdgcn_wmma_f32_16x16x4_f32(
                        false, a, false, bf, (short)0, c, false, false);
            }
        }

        // ================= Phase B: masked online softmax =================
        const int key  = T + wave * 16 + l15;     // this lane's key column
        const int mval = mg[key];
        float rmax[8];
#pragma unroll
        for (int r = 0; r < 8; ++r) {
            float v = (mval > 0) ? c[r] : NEG_BIG;
            c[r] = v;
            for (int mk = 1; mk <= 8; mk <<= 1)
                v = fmaxf(v, __shfl_xor(v, mk, 32));   // reduce within 16-lane half
            rmax[r] = v;
        }
        if (l15 == 0) {
#pragma unroll
            for (int r = 0; r < 8; ++r) wred[wave * 16 + r + 8 * half] = rmax[r];
        }
        __syncthreads();
        // all Phase A reads of Ks0/Ks1 are done: prefetch Phase C's V chunk 0
        if (wave == 0) tdm_load_tile(KS0_BYTE, Kblk);
        if (tid < 16) {
            float mb = wred[tid];
            for (int w = 1; w < 8; ++w) mb = fmaxf(mb, wred[w * 16 + tid]);
            float mo = mrun[tid];
            float mn = fmaxf(mo, mb);
            mnew[tid] = mn;
            scl[tid]  = __expf(mo - mn);
            mrun[tid] = mn;
        }
        __syncthreads();
        float rsum[8];
#pragma unroll
        for (int r = 0; r < 8; ++r) {
            float p = __expf(c[r] - mnew[r + 8 * half]);
            c[r] = p;
            float v = p;
            for (int mk = 1; mk <= 8; mk <<= 1)
                v += __shfl_xor(v, mk, 32);
            rsum[r] = v;
            Ps[(r + 8 * half) * PS_STRIDE + wave * 16 + l15] = p;
        }
        if (l15 == 0) {
#pragma unroll
            for (int r = 0; r < 8; ++r) wred[wave * 16 + r + 8 * half] = rsum[r];
        }
        __syncthreads();
        if (tid < 16) {
            float sb = 0.f;
            for (int w = 0; w < 8; ++w) sb += wred[w * 16 + tid];
            lrun[tid] = lrun[tid] * scl[tid] + sb;
        }

        // ================= Phase C: O = O*scale + P . V =================
        for (int hi = 0; hi < 8; ++hi) {
            const float* Vb = (hi & 1) ? Ks1 : Ks0;
            __builtin_amdgcn_s_wait_tensorcnt(0);
            __syncthreads();                       // V chunk hi ready, Ps/scl visible
            if (wave == 0 && hi < 7)               // overlap: fetch chunk hi+1
                tdm_load_tile((hi & 1) ? KS0_BYTE : KS1_BYTE,
                              Kblk + (hi + 1) * 128);
            v8f d = {0.f, 0.f, 0.f, 0.f, 0.f, 0.f, 0.f, 0.f};
            const int colbase = wave * 16;         // this wave's 16 output columns
            const float* prow = Ps + l15 * PS_STRIDE + 2 * half;
#pragma unroll
            for (int t4 = 0; t4 < 128; t4 += 4) {
                v2f a = *(const v2f*)(prow + t4);                // A = P[M=l15][K=t]
                v2f bf;
                bf.x = Vb[(t4 + 2 * half)     * KS_STRIDE + colbase + l15]; // V[t][h]
                bf.y = Vb[(t4 + 2 * half + 1) * KS_STRIDE + colbase + l15];
                d = __builtin_amdgcn_wmma_f32_16x16x4_f32(
                        false, a, false, bf, (short)0, d, false, false);
            }
            // read-modify-write O with the online-softmax rescale
#pragma unroll
            for (int r = 0; r < 8; ++r) {
                int row = r + 8 * half;
                int col = hi * 128 + colbase + l15;
                float o = Os[row * OS_STRIDE + col];
                Os[row * OS_STRIDE + col] = o * scl[row] + d[r];
            }
        }
        __syncthreads();
    }

    // ================= finalize: out = Q + O / l =================
    for (int i = 0; i < 16; ++i) {
        int r  = i;
        int c4 = tid;
        float invl = 1.0f / lrun[r];
        v4f q = *(const v4f*)(Qs + r * QS_STRIDE + c4 * 4);
        v4f o = *(const v4f*)(Os + r * OS_STRIDE + c4 * 4);
        v4f res;
        res.x = q.x + o.x * invl;
        res.y = q.y + o.y * invl;
        res.z = q.z + o.z * invl;
        res.w = q.w + o.w * invl;
        *(v4f*)(Og + (size_t)(qrow0 + r) * H_DIM + c4 * 4) = res;
    }
}

extern "C" void kernel_launch(void* const* d_in, const int* in_sizes, int n_in,
                              void* d_out, int out_size, void* d_ws, size_t ws_size,
                              hipStream_t stream) {
    const float* x    = (const float*)d_in[0];   // hidden_states (64,1024,1024) f32
    const int*   mask = (const int*)d_in[1];     // attention_mask (64,1024) i32
    float*       out  = (float*)d_out;

    (void)in_sizes; (void)n_in; (void)out_size; (void)d_ws; (void)ws_size;

    // dynamic LDS > 64KB needs the opt-in attribute (deterministic, capture-safe)
    hipFuncSetAttribute((const void*)fa_pairswap_kernel,
                        hipFuncAttributeMaxDynamicSharedMemorySize, LDS_BYTES);

    dim3 grid(S_DIM / 16, BATCH);   // 64 x 64 workgroups
    fa_pairswap_kernel<<<grid, dim3(256), LDS_BYTES, stream>>>(x, mask, out);
}